// AttentionModel_53249004536144
// MI455X (gfx1250) — compile-verified
//
#include <hip/hip_runtime.h>
#include <math.h>

typedef __attribute__((ext_vector_type(2))) float v2f;
typedef __attribute__((ext_vector_type(8))) float v8f;

#define H      128
#define B3H    384
#define NSTEP  2048   // INTER*F encoder steps
#define DECSTEP 512
#define ALEN   512    // MAX_LEN (attention width)
#define AVOC   512    // A (output vocab)

// workspace layout (float offsets)
#define WS_GI     0          // 2048*384  x-path GRU pre-activations (+enc_bih)
#define WS_EATTN  786432     // 512*512   dec_embed @ attn_W[:128] + attn_b
#define WS_ECOMB  1048576    // 512*128   dec_embed @ comb_W[:128] + comb_b
#define WS_ENCV   1114112    // 16*128    enc_vecs
#define WS_HFIN   1116160    // 128       final encoder hidden (row 0)

__device__ __forceinline__ float sigmoidf_(float x) { return 1.0f / (1.0f + expf(-x)); }

// ---------------------------------------------------------------------------
// Kernel 1: all precomputable GEMMs via fp32 WMMA (16x16x4), one wave per tile.
//   tiles 0..3071   : GI      (M=2048 gathered embeddings, N=384, K=128)
//   tiles 3072..4095: E_attn  (M=512 dec_embed rows,      N=512, K=128)
//   tiles 4096..4351: E_comb  (M=512 dec_embed rows,      N=128, K=128)
// A fragment: lane = (l&15 -> M), K = (l>>4)*2 + v   (16x4 f32, 2 VGPRs)
// B fragment: lane = (l&15 -> N), K = (l>>4)*2 + v   (4x16 f32, 2 VGPRs)
// C fragment: VGPR r -> M = r + 8*(l>>4), N = l&15   (8 VGPRs)
// ---------------------------------------------------------------------------
__global__ void tables_wmma(const int* __restrict__ obs,
                            const float* __restrict__ enc_embed,
                            const float* __restrict__ enc_Wih,
                            const float* __restrict__ enc_bih,
                            const float* __restrict__ dec_embed,
                            const float* __restrict__ attn_W,
                            const float* __restrict__ attn_b,
                            const float* __restrict__ comb_W,
                            const float* __restrict__ comb_b,
                            float* __restrict__ ws)
{
    const int lane = threadIdx.x & 31;
    const int wid  = (blockIdx.x * blockDim.x + threadIdx.x) >> 5;
    const int mrow  = lane & 15;        // A row within tile == C column lane
    const int khalf = (lane >> 4) << 1; // 0 or 2 (K sub-pair per half-wave)

    const float* Arow; const float* Bmat; const float* bias; float* C;
    int ldb, ldc, mbase, nbase;

    if (wid < 3072) {               // GI = X @ enc_Wih + enc_bih
        int mt = wid / 24, nt = wid % 24;
        mbase = mt * 16; nbase = nt * 16;
        int t   = mbase + mrow;
        // emb_seq[t, 0] = enc_embed[obs_flat[(t>>7)*65536 + (t&127)]]
        int idx = obs[((t >> 7) << 16) | (t & 127)];
        Arow = enc_embed + idx * H;
        Bmat = enc_Wih; ldb = B3H; bias = enc_bih;
        C = ws + WS_GI; ldc = B3H;
    } else if (wid < 4096) {        // E_attn = dec_embed @ attn_W[:128] + attn_b
        int id = wid - 3072; int mt = id >> 5, nt = id & 31;
        mbase = mt * 16; nbase = nt * 16;
        Arow = dec_embed + (mbase + mrow) * H;
        Bmat = attn_W; ldb = ALEN; bias = attn_b;
        C = ws + WS_EATTN; ldc = ALEN;
    } else if (wid < 4352) {        // E_comb = dec_embed @ comb_W[:128] + comb_b
        int id = wid - 4096; int mt = id >> 3, nt = id & 7;
        mbase = mt * 16; nbase = nt * 16;
        Arow = dec_embed + (mbase + mrow) * H;
        Bmat = comb_W; ldb = H; bias = comb_b;
        C = ws + WS_ECOMB; ldc = H;
    } else {
        return;
    }

    const int col = nbase + mrow;
    v8f c;
    const float bv = bias[col];     // bias depends only on N -> same for all 8 rows
    #pragma unroll
    for (int r = 0; r < 8; ++r) c[r] = bv;

    #pragma unroll                  // full unroll: 32 static v_wmma, deep load pipelining
    for (int k0 = 0; k0 < H; k0 += 4) {
        v2f a = *(const v2f*)(Arow + k0 + khalf);   // 8B-aligned contiguous pair
        v2f b;
        b.x = Bmat[(k0 + khalf)     * ldb + col];
        b.y = Bmat[(k0 + khalf + 1) * ldb + col];
        c = __builtin_amdgcn_wmma_f32_16x16x4_f32(false, a, false, b,
                                                  (short)0, c, false, false);
    }

    const int rbase = mbase + ((lane >> 4) << 3);
    float* Cp = C + rbase * ldc + col;              // 32-bit index math only
    #pragma unroll
    for (int r = 0; r < 8; ++r) Cp[r * ldc] = c[r];
}

// ---------------------------------------------------------------------------
// Kernel 2: sequential encoder (batch row 0 only), single workgroup.
// enc_Whh (192KB fp32) resident in LDS; GI rows streamed from L2.
// ---------------------------------------------------------------------------
#define ENC_THREADS 512
__global__ void encoder_seq(const float* __restrict__ enc_Whh,
                            const float* __restrict__ enc_bhh,
                            float* __restrict__ ws)
{
    extern __shared__ float lds[];
    float* Wl  = lds;            // 128*384
    float* hb  = Wl + H * B3H;   // 128
    float* gh  = hb + H;         // 384
    float* bhh = gh + B3H;       // 384
    const float* GI = ws + WS_GI;
    float* encv = ws + WS_ENCV;
    float* hfin = ws + WS_HFIN;
    const int tid = threadIdx.x;

    for (int i = tid; i < H * B3H; i += ENC_THREADS) Wl[i] = enc_Whh[i];
    if (tid < B3H) bhh[tid] = enc_bhh[tid];
    if (tid < H)   hb[tid]  = 0.0f;          // h0 = zeros
    __syncthreads();

    for (int t = 0; t < NSTEP; ++t) {
        if (tid < B3H) {                     // gh = h @ Whh + bhh
            float acc = bhh[tid];
            #pragma unroll 8
            for (int k = 0; k < H; ++k) acc += hb[k] * Wl[k * B3H + tid];
            gh[tid] = acc;
        }
        if (tid < B3H && t + 1 < NSTEP)      // warm caches for next GI row
            __builtin_prefetch(&GI[(size_t)(t + 1) * B3H + tid], 0, 0);
        __syncthreads();
        if (tid < H) {
            const float* gi = GI + (size_t)t * B3H;
            float r  = sigmoidf_(gi[tid]         + gh[tid]);
            float z  = sigmoidf_(gi[H + tid]     + gh[H + tid]);
            float n  = tanhf    (gi[2 * H + tid] + r * gh[2 * H + tid]);
            float hn = (1.0f - z) * n + z * hb[tid];
            if ((t & 127) == 0) encv[(t >> 7) * H + tid] = hn;  // enc_outs[k*F, 0]
            hb[tid] = hn;
        }
        __syncthreads();
    }
    if (tid < H) hfin[tid] = hb[tid];
}

// ---------------------------------------------------------------------------
// Kernel 3: sequential decoder (batch row 0 only), single workgroup.
// attn_W[128:] (256KB) in LDS; EV_comb table built in prologue; the
// score-matvec and gh-matvec run concurrently on disjoint thread ranges.
// ---------------------------------------------------------------------------
#define DEC_THREADS 1024
__global__ void decoder_seq(const float* __restrict__ attn_W,
                            const float* __restrict__ comb_W,
                            const float* __restrict__ dec_Wih,
                            const float* __restrict__ dec_Whh,
                            const float* __restrict__ dec_bih,
                            const float* __restrict__ dec_bhh,
                            const float* __restrict__ out_W,
                            const float* __restrict__ out_b,
                            const float* __restrict__ ws,
                            float* __restrict__ out)
{
    extern __shared__ float lds[];
    float* aWb  = lds;                 // 128*512  attn_W rows 128..255
    float* EVc  = aWb + H * ALEN;      // 16*128   enc_vecs @ comb_W[128:]
    float* sc   = EVc + 16 * H;        // 512      scores / weights / logits
    float* gh   = sc + ALEN;           // 384
    float* gi   = gh + B3H;            // 384
    float* bih  = gi + B3H;            // 384
    float* bhh  = bih + B3H;           // 384
    float* outb = bhh + B3H;           // 512
    float* hb   = outb + AVOC;         // 128
    float* hn   = hb + H;              // 128
    float* ob   = hn + H;              // 128
    float* red  = ob + H;              // 64
    float* sval = red + 64;            // [0]=max, [1]=sum/logsum
    int*   redi = (int*)(sval + 2);    // 64
    int*   sinp = redi + 64;           // fed-back argmax token

    const int tid = threadIdx.x;
    const float* E_attn = ws + WS_EATTN;
    const float* E_comb = ws + WS_ECOMB;
    const float* encv   = ws + WS_ENCV;

    // ---- prologue ----
    for (int i = tid; i < H * ALEN; i += DEC_THREADS)
        aWb[i] = attn_W[(H + (i >> 9)) * ALEN + (i & 511)];
    for (int i = tid; i < 16 * H; i += DEC_THREADS) {
        int k = i >> 7, j = i & 127;
        float acc = 0.0f;
        for (int c = 0; c < H; ++c) acc += encv[k * H + c] * comb_W[(H + c) * H + j];
        EVc[i] = acc;
    }
    if (tid < B3H) { bih[tid] = dec_bih[tid]; bhh[tid] = dec_bhh[tid]; }
    if (tid < AVOC) outb[tid] = out_b[tid];
    if (tid < H)    hb[tid]   = ws[WS_HFIN + tid];
    if (tid == 0)   *sinp = 0;
    __syncthreads();

    for (int t = 0; t < DECSTEP; ++t) {
        const int inp = *sinp;
        // A: attention scores (tid<512) || gh = h @ dec_Whh (512<=tid<896)
        if (tid < ALEN) {
            float acc = E_attn[inp * ALEN + tid];
            #pragma unroll 8
            for (int k = 0; k < H; ++k) acc += hb[k] * aWb[k * ALEN + tid];
            sc[tid] = acc;
        } else if (tid < ALEN + B3H) {
            const int j = tid - ALEN;
            float acc = bhh[j];
            #pragma unroll 4
            for (int k = 0; k < H; ++k) acc += hb[k] * dec_Whh[k * B3H + j];
            gh[j] = acc;
        }
        __syncthreads();
        // B: softmax over scores
        if (tid < 64) {
            float m = sc[tid * 8];
            for (int i = 1; i < 8; ++i) m = fmaxf(m, sc[tid * 8 + i]);
            red[tid] = m;
        }
        __syncthreads();
        if (tid == 0) {
            float m = red[0];
            for (int i = 1; i < 64; ++i) m = fmaxf(m, red[i]);
            sval[0] = m;
        }
        __syncthreads();
        const float amx = sval[0];
        if (tid < ALEN) sc[tid] = expf(sc[tid] - amx);
        __syncthreads();
        if (tid < 64) {
            float s = 0.0f;
            for (int i = 0; i < 8; ++i) s += sc[tid * 8 + i];
            red[tid] = s;
        }
        __syncthreads();
        if (tid == 0) {
            float s = 0.0f;
            for (int i = 0; i < 64; ++i) s += red[i];
            sval[1] = s;
        }
        __syncthreads();
        // C: combine (only 16 nonzero encoder_outputs rows) + ReLU
        const float invden = 1.0f / sval[1];
        if (tid < H) {
            float acc = E_comb[inp * H + tid];
            #pragma unroll
            for (int k = 0; k < 16; ++k) acc += (sc[k] * invden) * EVc[k * H + tid];
            ob[tid] = fmaxf(acc, 0.0f);
        }
        __syncthreads();
        // D: gi = o @ dec_Wih + bih
        if (tid < B3H) {
            float acc = bih[tid];
            #pragma unroll 4
            for (int k = 0; k < H; ++k) acc += ob[k] * dec_Wih[k * B3H + tid];
            gi[tid] = acc;
        }
        __syncthreads();
        // E: GRU gates
        if (tid < H) {
            float r = sigmoidf_(gi[tid]         + gh[tid]);
            float z = sigmoidf_(gi[H + tid]     + gh[H + tid]);
            float n = tanhf    (gi[2 * H + tid] + r * gh[2 * H + tid]);
            hn[tid] = (1.0f - z) * n + z * hb[tid];
        }
        __syncthreads();
        // F: logits = h_new @ out_W + out_b  (||  h <- h_new on spare threads)
        if (tid < AVOC) {
            float acc = outb[tid];
            #pragma unroll 8
            for (int k = 0; k < H; ++k) acc += hn[k] * out_W[k * AVOC + tid];
            sc[tid] = acc;
        } else if (tid < AVOC + H) {
            hb[tid - AVOC] = hn[tid - AVOC];
        }
        __syncthreads();
        // G: argmax (first-occurrence, matches jnp.argmax) + log_softmax
        if (tid < 64) {
            float m = sc[tid * 8]; int mi = tid * 8;
            for (int i = 1; i < 8; ++i) {
                float v = sc[tid * 8 + i];
                if (v > m) { m = v; mi = tid * 8 + i; }
            }
            red[tid] = m; redi[tid] = mi;
        }
        __syncthreads();
        if (tid == 0) {
            float m = red[0]; int mi = redi[0];
            for (int i = 1; i < 64; ++i)
                if (red[i] > m) { m = red[i]; mi = redi[i]; }
            sval[0] = m; *sinp = mi;
        }
        __syncthreads();
        const float lmx = sval[0];
        if (tid < 64) {
            float s = 0.0f;
            for (int i = 0; i < 8; ++i) s += expf(sc[tid * 8 + i] - lmx);
            red[tid] = s;
        }
        __syncthreads();
        if (tid == 0) {
            float s = 0.0f;
            for (int i = 0; i < 64; ++i) s += red[i];
            sval[1] = logf(s);
        }
        __syncthreads();
        if (tid < AVOC) out[(size_t)t * AVOC + tid] = sc[tid] - lmx - sval[1];
        __syncthreads();
    }
}

// ---------------------------------------------------------------------------
extern "C" void kernel_launch(void* const* d_in, const int* in_sizes, int n_in,
                              void* d_out, int out_size, void* d_ws, size_t ws_size,
                              hipStream_t stream) {
    (void)in_sizes; (void)n_in; (void)out_size; (void)ws_size;
    const int*   obs       = (const int*)  d_in[0];
    const float* enc_embed = (const float*)d_in[1];
    const float* enc_Wih   = (const float*)d_in[2];
    const float* enc_Whh   = (const float*)d_in[3];
    const float* enc_bih   = (const float*)d_in[4];
    const float* enc_bhh   = (const float*)d_in[5];
    const float* dec_embed = (const float*)d_in[6];
    const float* attn_W    = (const float*)d_in[7];
    const float* attn_b    = (const float*)d_in[8];
    const float* comb_W    = (const float*)d_in[9];
    const float* comb_b    = (const float*)d_in[10];
    const float* dec_Wih   = (const float*)d_in[11];
    const float* dec_Whh   = (const float*)d_in[12];
    const float* dec_bih   = (const float*)d_in[13];
    const float* dec_bhh   = (const float*)d_in[14];
    const float* out_W     = (const float*)d_in[15];
    const float* out_b     = (const float*)d_in[16];
    float* ws  = (float*)d_ws;
    float* out = (float*)d_out;

    const size_t enc_lds = (size_t)(H * B3H + H + B3H + B3H) * sizeof(float); // ~196KB
    const size_t dec_lds = (size_t)(H * ALEN + 16 * H + ALEN + 4 * B3H + AVOC
                                    + 3 * H + 64 + 2 + 65) * sizeof(float);   // ~276KB
    hipFuncSetAttribute((const void*)encoder_seq,
                        hipFuncAttributeMaxDynamicSharedMemorySize, (int)enc_lds);
    hipFuncSetAttribute((const void*)decoder_seq,
                        hipFuncAttributeMaxDynamicSharedMemorySize, (int)dec_lds);

    // 4352 WMMA tiles, 8 waves per 256-thread block -> 544 blocks
    tables_wmma<<<544, 256, 0, stream>>>(obs, enc_embed, enc_Wih, enc_bih,
                                         dec_embed, attn_W, attn_b,
                                         comb_W, comb_b, ws);
    encoder_seq<<<1, ENC_THREADS, enc_lds, stream>>>(enc_Whh, enc_bhh, ws);
    decoder_seq<<<1, DEC_THREADS, dec_lds, stream>>>(attn_W, comb_W, dec_Wih, dec_Whh,
                                                     dec_bih, dec_bhh, out_W, out_b,
                                                     ws, out);
}